// FullMatrix_61942018342933
// MI455X (gfx1250) — compile-verified
//
#include <hip/hip_runtime.h>

// ---------------------------------------------------------------------------
// Problem constants (reference: D=128, B=4, S=512)
// ---------------------------------------------------------------------------
#define DIM   128
#define BATCH 4
#define SEQ   512
#define NTOK  (BATCH * SEQ)   // 2048 tokens
#define EDIM  (DIM * DIM)     // 16384 projection outputs per token

typedef __attribute__((ext_vector_type(2))) float v2f;
typedef __attribute__((ext_vector_type(8))) float v8f;
typedef unsigned int v4u __attribute__((ext_vector_type(4)));
typedef int v4i __attribute__((ext_vector_type(4)));
typedef int v8i __attribute__((ext_vector_type(8)));

// Full-precision fp32 matrix op: V_WMMA_F32_16X16X4_F32 (D = A(16x4) * B(4x16) + C)
__device__ __forceinline__ v8f wmma_f32_k4(v2f a, v2f b, v8f c) {
    return __builtin_amdgcn_wmma_f32_16x16x4_f32(
        /*neg_a=*/false, a, /*neg_b=*/false, b,
        /*c_mod=*/(short)0, c, /*reuse_a=*/false, /*reuse_b=*/false);
}

__device__ __forceinline__ v8f zero_v8f() {
    v8f z;
#pragma unroll
    for (int q = 0; q < 8; ++q) z[q] = 0.0f;
    return z;
}

#define INV_SQRT_D 0.08838834764831845f  // 1/sqrt(128)

// ---------------------------------------------------------------------------
// Kernel 1: A projection GEMM.  Aout[token][e] = (x[token,:] . W[e,:] + b[e])/sqrt(D)
// M = 2048 tokens, N = 16384, K = 128 via V_WMMA_F32_16X16X4_F32.
// Aout (128 MiB) stays resident in the 192MB L2 for the scan kernel.
// ---------------------------------------------------------------------------
__global__ __launch_bounds__(256) void proj_A_kernel(
    const float* __restrict__ x,     // [NTOK, DIM]
    const float* __restrict__ W,     // [EDIM, DIM]
    const float* __restrict__ bias,  // [EDIM]
    float* __restrict__ Aout)        // [NTOK, EDIM]
{
    const int lane = threadIdx.x & 31;
    const int wave = threadIdx.x >> 5;
    const int hf   = lane >> 4;   // 0: lanes 0-15 (K=0,1), 1: lanes 16-31 (K=2,3)
    const int m    = lane & 15;   // row (A) / col (B,D) within tile

    const int tokBase = blockIdx.x * 16;                 // 128 token tiles
    const int eBase0  = blockIdx.y * 512 + wave * 64;    // 32 e-groups of 512

    const float* xrow = x + (size_t)(tokBase + m) * DIM;

    v8f acc[4];
#pragma unroll
    for (int t = 0; t < 4; ++t) acc[t] = zero_v8f();

    for (int k = 0; k < DIM; k += 4) {
        v2f a = *(const v2f*)(xrow + k + 2 * hf);
#pragma unroll
        for (int t = 0; t < 4; ++t) {
            const float* wrow = W + (size_t)(eBase0 + t * 16 + m) * DIM;
            v2f b = *(const v2f*)(wrow + k + 2 * hf);
            acc[t] = wmma_f32_k4(a, b, acc[t]);
        }
    }

#pragma unroll
    for (int t = 0; t < 4; ++t) {
        const int e  = eBase0 + t * 16 + m;
        const float bv = bias[e];
#pragma unroll
        for (int r = 0; r < 8; ++r) {
            const int tok = tokBase + r + 8 * hf;
            Aout[(size_t)tok * EDIM + e] = (acc[t][r] + bv) * INV_SQRT_D;
        }
    }
}

// ---------------------------------------------------------------------------
// Kernel 2: Bx projection, fused (never materializes Bm).
// ---------------------------------------------------------------------------
__global__ __launch_bounds__(256) void proj_Bx_kernel(
    const float* __restrict__ x,     // [NTOK, DIM]
    const float* __restrict__ W,     // [EDIM, DIM]
    const float* __restrict__ bias,  // [EDIM]
    float* __restrict__ Bx)          // [NTOK, DIM]
{
    const int lane = threadIdx.x & 31;
    const int wave = threadIdx.x >> 5;
    const int hf   = lane >> 4;
    const int m    = lane & 15;

    const int tokBase = blockIdx.x * 16;        // 128 token tiles
    const int i       = blockIdx.y * 8 + wave;  // output row index 0..127

    const float* xrow = x + (size_t)(tokBase + m) * DIM;

    v8f acc[8];
#pragma unroll
    for (int t = 0; t < 8; ++t) acc[t] = zero_v8f();

    for (int k = 0; k < DIM; k += 4) {
        v2f a = *(const v2f*)(xrow + k + 2 * hf);
#pragma unroll
        for (int jt = 0; jt < 8; ++jt) {
            const float* wrow = W + (size_t)(i * DIM + jt * 16 + m) * DIM;
            v2f b = *(const v2f*)(wrow + k + 2 * hf);
            acc[jt] = wmma_f32_k4(a, b, acc[jt]);
        }
    }

    float p[8];
#pragma unroll
    for (int r = 0; r < 8; ++r) p[r] = 0.0f;
#pragma unroll
    for (int jt = 0; jt < 8; ++jt) {
        const int j  = jt * 16 + m;
        const float bv = bias[i * DIM + j];
#pragma unroll
        for (int r = 0; r < 8; ++r) {
            const int tok = tokBase + r + 8 * hf;
            p[r] += (acc[jt][r] + bv) * x[(size_t)tok * DIM + j];
        }
    }

#pragma unroll
    for (int r = 0; r < 8; ++r) {
        float v = p[r];
        v += __shfl_xor(v, 1, 16);
        v += __shfl_xor(v, 2, 16);
        v += __shfl_xor(v, 4, 16);
        v += __shfl_xor(v, 8, 16);
        if (m == 0) {
            const int tok = tokBase + r + 8 * hf;
            Bx[(size_t)tok * DIM + i] = v * INV_SQRT_D;
        }
    }
}

// ---------------------------------------------------------------------------
// TDM: DMA one 64KB A_t tile (16384 f32, contiguous) from global into LDS.
// D# per ISA cdna5 §8: group0 = {count=1, lds_addr, global_addr(57b), type=2},
// group1 = {data_size=4B, tensor_dim0=EDIM, tensor_dim1=SEQ, tile_dim0=EDIM,
//           tile_dim1=1, dim0_stride=EDIM}; groups 2/3 zero (<=2D tensor).
// This toolchain exposes the 6-arg builtin (extra int32x8 descriptor group).
// Issued by one wave per workgroup; tracked by TENSORcnt.
// ---------------------------------------------------------------------------
#if __has_builtin(__builtin_amdgcn_tensor_load_to_lds) && \
    __has_builtin(__builtin_amdgcn_s_wait_tensorcnt)
#define USE_TDM 1
#else
#define USE_TDM 0
#endif

#if USE_TDM
__device__ __forceinline__ void tdm_load_tile(const float* gsrc, unsigned int lds_off) {
    const unsigned long long ga = (unsigned long long)(size_t)gsrc;
    v4u g0;
    g0[0] = 1u;                                               // count=1, no gather
    g0[1] = lds_off;                                          // lds_addr (bytes)
    g0[2] = (unsigned int)ga;                                 // global_addr[31:0]
    g0[3] = ((unsigned int)(ga >> 32) & 0x01FFFFFFu)          // global_addr[56:32]
          | 0x80000000u;                                      // type=2 ("image")
    v8i g1;
    g1[0] = 0x00020000;                                       // data_size=4B; mask=0
    g1[1] = (int)((EDIM & 0xFFFF) << 16);                     // tensor_dim0[15:0]
    g1[2] = (int)((EDIM >> 16) | ((SEQ & 0xFFFF) << 16));     // dim0[31:16] | dim1[15:0]
    g1[3] = (int)((EDIM & 0xFFFF) << 16);                     // dim1[31:16]=0 | tile_dim0
    g1[4] = 1;                                                // tile_dim1=1, tile_dim2=0
    g1[5] = EDIM;                                             // dim0_stride[31:0]
    g1[6] = 0;                                                // stride hi | dim1_stride lo
    g1[7] = 0;
    v4i g2 = {0, 0, 0, 0};                                    // group2: unused (<=2D)
    v4i g3 = {0, 0, 0, 0};                                    // group3: unused
    v8i g4 = {0, 0, 0, 0, 0, 0, 0, 0};                        // extra group (6-arg form)
    __builtin_amdgcn_tensor_load_to_lds(g0, g1, g2, g3, g4, 0);
}
#endif

// ---------------------------------------------------------------------------
// Kernel 3: sequential scan  h_s = A_s h_{s-1} + Bx_s, one workgroup per batch.
// 512 threads: thread t owns (i = t/4, j-chunk = t%4 of 32 columns); h in LDS.
// A_{s+1} is DMA'd by the Tensor Data Mover into an LDS double buffer while the
// VALU computes step s — zero vector-issue cost on the 512-step serial chain.
// Dynamic LDS: 2 x 64KB tile buffers + h ping-pong + reduction scratch.
// ---------------------------------------------------------------------------
__global__ __launch_bounds__(512) void scan_kernel(
    const float* __restrict__ Amat,  // [NTOK, EDIM]  (already biased + scaled)
    const float* __restrict__ Bx,    // [NTOK, DIM]
    float* __restrict__ out)         // [NTOK, DIM]
{
    extern __shared__ float smem[];
    float* Atile0 = smem;                       // EDIM floats (64 KB)
    float* Atile1 = smem + EDIM;                // EDIM floats (64 KB)
    float* hb     = smem + 2 * EDIM;            // 2*DIM floats (h ping-pong)
    float* red    = smem + 2 * EDIM + 2 * DIM;  // 512 floats

    const int tid = threadIdx.x;
    const int i   = tid >> 2;   // 0..127 output row
    const int jc  = tid & 3;    // 0..3  column chunk (32 cols)
    const int b   = blockIdx.x;

    const float* Ab  = Amat + (size_t)b * SEQ * EDIM;
    const float* Bxb = Bx   + (size_t)b * SEQ * DIM;
    float*       ob  = out  + (size_t)b * SEQ * DIM;

    if (tid < DIM) hb[tid] = 0.0f;   // h_{-1} = 0

#if USE_TDM
    const unsigned int lds0 = (unsigned int)(size_t)Atile0;
    const unsigned int lds1 = (unsigned int)(size_t)Atile1;
    if (tid < 32) {                       // wave 0 issues the DMA (per-wave op)
        tdm_load_tile(Ab, lds0);          // tile for s=0
        __builtin_amdgcn_s_wait_tensorcnt(0);
    }
    __syncthreads();                      // tile 0 + h visible to all waves

    int cur = 0;
    for (int s = 0; s < SEQ; ++s) {
        // Kick off DMA of A_{s+1} into the other buffer; overlaps compute below.
        if (tid < 32 && (s + 1) < SEQ)
            tdm_load_tile(Ab + (size_t)(s + 1) * EDIM, cur ? lds0 : lds1);

        const float* arow = (cur ? Atile1 : Atile0) + i * DIM + jc * 32;
        const float* h    = hb + ((s & 1) ? DIM : 0);
        float partial = 0.0f;
#pragma unroll
        for (int q = 0; q < 8; ++q) {
            const float4 a = ((const float4*)arow)[q];
            const int j = jc * 32 + q * 4;
            partial += a.x * h[j + 0] + a.y * h[j + 1]
                     + a.z * h[j + 2] + a.w * h[j + 3];
        }
        red[tid] = partial;
        __syncthreads();
        if (jc == 0) {
            const float hn = red[i * 4 + 0] + red[i * 4 + 1]
                           + red[i * 4 + 2] + red[i * 4 + 3] + Bxb[s * DIM + i];
            hb[((s & 1) ? 0 : DIM) + i] = hn;   // next-step h buffer
            ob[s * DIM + i] = hn;
        }
        if (tid < 32)
            __builtin_amdgcn_s_wait_tensorcnt(0);  // next tile landed in LDS
        __syncthreads();                           // publish h + tile to all waves
        cur ^= 1;
    }
#else
    __syncthreads();
    // Fallback: register double-buffer prefetch (no TDM builtin available).
    float4 areg[8];
    {
        const float4* src = (const float4*)(Ab + (size_t)i * DIM + jc * 32);
#pragma unroll
        for (int q = 0; q < 8; ++q) areg[q] = src[q];
    }
    for (int s = 0; s < SEQ; ++s) {
        float4 anext[8];
        if (s + 1 < SEQ) {
            const float4* src = (const float4*)(Ab + (size_t)(s + 1) * EDIM
                                                + (size_t)i * DIM + jc * 32);
#pragma unroll
            for (int q = 0; q < 8; ++q) anext[q] = src[q];
        } else {
#pragma unroll
            for (int q = 0; q < 8; ++q) anext[q] = areg[q];
        }
        const float* h = hb + ((s & 1) ? DIM : 0);
        float partial = 0.0f;
#pragma unroll
        for (int q = 0; q < 8; ++q) {
            const int j = jc * 32 + q * 4;
            partial += areg[q].x * h[j + 0] + areg[q].y * h[j + 1]
                     + areg[q].z * h[j + 2] + areg[q].w * h[j + 3];
        }
        red[tid] = partial;
        __syncthreads();
        if (jc == 0) {
            const float hn = red[i * 4 + 0] + red[i * 4 + 1]
                           + red[i * 4 + 2] + red[i * 4 + 3] + Bxb[s * DIM + i];
            hb[((s & 1) ? 0 : DIM) + i] = hn;
            ob[s * DIM + i] = hn;
        }
        __syncthreads();
#pragma unroll
        for (int q = 0; q < 8; ++q) areg[q] = anext[q];
    }
#endif
}

// ---------------------------------------------------------------------------
// Launch: proj_A and proj_Bx are independent; scan consumes both.
// Workspace: A matrices (128 MiB, L2-resident) + Bx (1 MiB).
// ---------------------------------------------------------------------------
extern "C" void kernel_launch(void* const* d_in, const int* in_sizes, int n_in,
                              void* d_out, int out_size, void* d_ws, size_t ws_size,
                              hipStream_t stream) {
    const float* x   = (const float*)d_in[0];
    const float* A_w = (const float*)d_in[1];
    const float* A_b = (const float*)d_in[2];
    const float* B_w = (const float*)d_in[3];
    const float* B_b = (const float*)d_in[4];
    float* out = (float*)d_out;

    float* A_ws  = (float*)d_ws;                     // NTOK*EDIM floats (128 MiB)
    float* Bx_ws = A_ws + (size_t)NTOK * EDIM;       // NTOK*DIM floats  (1 MiB)

    dim3 blk(256);
    proj_A_kernel <<<dim3(NTOK / 16, EDIM / 512), blk, 0, stream>>>(x, A_w, A_b, A_ws);
    proj_Bx_kernel<<<dim3(NTOK / 16, DIM / 8),    blk, 0, stream>>>(x, B_w, B_b, Bx_ws);

    const size_t smem_bytes = (2 * (size_t)EDIM + 2 * DIM + 512) * sizeof(float);
    scan_kernel<<<dim3(BATCH), dim3(512), smem_bytes, stream>>>(A_ws, Bx_ws, out);
}